// MixGAT_14697378087235
// MI455X (gfx1250) — compile-verified
//
#include <hip/hip_runtime.h>
#include <cstdint>

#define SLOPE  0.2f
#define BETA   0.5f
#define CCONST 1.0f
#define GEMM_KMAX 256   // K is 256 for both layers (IN=256, H*HID=256)

typedef float v2f __attribute__((ext_vector_type(2)));
typedef float v8f __attribute__((ext_vector_type(8)));

// ---------------------------------------------------------------------------
// FP32 WMMA GEMM: C[M x Nc] = A[M x K] * B[K x Nc], row-major.
// Requires M%16==0, K%8==0, K<=GEMM_KMAX, Nc%64==0.
// 8-wave workgroup computes a 128(M) x 64(N) tile. The K x 64 strip of B is
// staged once into LDS in K-pair-interleaved layout:
//     Bs[(k/2)*128 + 2*n + (k&1)] = B[k][tn+n]
// so each lane's WMMA B operand {B[ka][n], B[ka+1][n]} is ONE aligned
// ds_load_b64 — no register re-pairing in the inner loop.
// ---------------------------------------------------------------------------
__device__ __forceinline__ v8f wmma4(v2f a, v2f b, v8f c) {
  return __builtin_amdgcn_wmma_f32_16x16x4_f32(
      /*neg_a=*/false, a, /*neg_b=*/false, b,
      /*c_mod=*/(short)0, c, /*reuse_a=*/false, /*reuse_b=*/false);
}

__global__ void gat_gemm_wmma(const float* __restrict__ A, const float* __restrict__ B,
                              float* __restrict__ C, int M, int K, int Nc) {
  __shared__ float Bs[GEMM_KMAX * 64];  // (K/2) x 128, pair-interleaved
  const int tid = threadIdx.x;          // 0..255
  const int tn  = blockIdx.x * 64;

  // Stage B -> LDS. Each idx handles one (k-pair, 4-column) group:
  // two b128 global row loads, four b64 interleaved LDS stores.
  const int npair = (K >> 1) * 16;
  for (int idx = tid; idx < npair; idx += 256) {
    const int kp = idx >> 4;
    const int n4 = (idx & 15) << 2;
    const float4 r0 = *(const float4*)&B[(size_t)(2 * kp + 0) * Nc + tn + n4];
    const float4 r1 = *(const float4*)&B[(size_t)(2 * kp + 1) * Nc + tn + n4];
    float* dst = &Bs[kp * 128 + 2 * n4];
    *(float2*)(dst + 0) = make_float2(r0.x, r1.x);
    *(float2*)(dst + 2) = make_float2(r0.y, r1.y);
    *(float2*)(dst + 4) = make_float2(r0.z, r1.z);
    *(float2*)(dst + 6) = make_float2(r0.w, r1.w);
  }
  __syncthreads();

  const int wave = tid >> 5;
  const int lane = tid & 31;
  const int l15  = lane & 15;           // M (A) / N (B,C) index within tile
  const int half = lane >> 4;           // selects K pair {0,1} vs {2,3}
  const int tm   = (blockIdx.y * 8 + wave) * 16;
  if (tm >= M) return;                  // tail waves: staged + barriered, done

  const float* Arow = A + (size_t)(tm + l15) * K;
  v8f acc0 = {}, acc1 = {}, acc2 = {}, acc3 = {};
  for (int k = 0; k < K; k += 4) {
    const int ka = k + half * 2;
    v2f a; a.x = Arow[ka]; a.y = Arow[ka + 1];
    const float* Bp = &Bs[(ka >> 1) * 128 + 2 * l15];
    acc0 = wmma4(a, *(const v2f*)(Bp + 0),  acc0);
    acc1 = wmma4(a, *(const v2f*)(Bp + 32), acc1);
    acc2 = wmma4(a, *(const v2f*)(Bp + 64), acc2);
    acc3 = wmma4(a, *(const v2f*)(Bp + 96), acc3);
  }
  for (int r = 0; r < 8; ++r) {
    float* Crow = C + (size_t)(tm + half * 8 + r) * Nc + tn + l15;
    Crow[0]  = acc0[r];
    Crow[16] = acc1[r];
    Crow[32] = acc2[r];
    Crow[48] = acc3[r];
  }
}

// ---------------------------------------------------------------------------
// Elementwise helpers
// ---------------------------------------------------------------------------
__global__ void gat_fill(float* p, long n, float v) {
  long i = (long)blockIdx.x * blockDim.x + threadIdx.x;
  if (i < n) p[i] = v;
}

__global__ void gat_bias_init(float* p, long n, int C, const float* __restrict__ b) {
  long i = (long)blockIdx.x * blockDim.x + threadIdx.x;
  if (i < n) p[i] = b[(int)(i % C)];
}

// x1 = BETA*z + (C-BETA)*swish(z), in place (z already includes bias from init)
__global__ void gat_mix(float* p, long n) {
  long i = (long)blockIdx.x * blockDim.x + threadIdx.x;
  if (i >= n) return;
  float z  = p[i];
  float sg = 1.0f / (1.0f + __expf(-z));
  p[i] = BETA * z + (CCONST - BETA) * (z * sg);
}

__global__ void gat_scale(float* p, long n, float s) {
  long i = (long)blockIdx.x * blockDim.x + threadIdx.x;
  if (i < n) p[i] *= s;
}

// ---------------------------------------------------------------------------
// Attention logits: s_src[n,h]=dot(h[n,h,:],a_src[h,:]); same for s_dst.
// One thread per (n,h). feat layout [n][h][f] => flat index i*F works.
// ---------------------------------------------------------------------------
__global__ void gat_logits(const float* __restrict__ feat,
                           const float* __restrict__ a_src,
                           const float* __restrict__ a_dst,
                           float* __restrict__ ss, float* __restrict__ sd,
                           long NH, int Hh, int F) {
  long i = (long)blockIdx.x * blockDim.x + threadIdx.x;
  if (i >= NH) return;
  int h0 = (int)(i % Hh);
  const float* fp = feat + (size_t)i * F;
  const float* as = a_src + (size_t)h0 * F;
  const float* ad = a_dst + (size_t)h0 * F;
  float vs = 0.f, vd = 0.f;
  for (int f = 0; f < F; ++f) { float v = fp[f]; vs += v * as[f]; vd += v * ad[f]; }
  ss[i] = vs; sd[i] = vd;
}

// ---------------------------------------------------------------------------
// Edge passes. Edge e < E comes from edge_index; e >= E is self-loop (e-E).
// ---------------------------------------------------------------------------
__device__ __forceinline__ void gat_endpoints(const int* __restrict__ ei, int E,
                                              long e, int& s, int& d) {
  if (e < E) { s = ei[e]; d = ei[(size_t)E + e]; }
  else       { s = d = (int)(e - E); }
}

__device__ __forceinline__ float gat_leaky(float v) {
  return v > 0.f ? v : SLOPE * v;
}

// float atomic max via monotonic ordered-int-bits mapping (init must be -inf)
__device__ __forceinline__ void gat_atomic_max_f32(float* addr, float val) {
  int iv = __float_as_int(val);
  if (iv >= 0) atomicMax((int*)addr, iv);
  else         atomicMin((unsigned int*)addr, (unsigned int)iv);
}

__global__ void gat_edge_max(const int* __restrict__ ei, int E, int N, int Hh,
                             const float* __restrict__ ss, const float* __restrict__ sd,
                             float* __restrict__ emax) {
  long i = (long)blockIdx.x * blockDim.x + threadIdx.x;
  long total = (long)(E + N) * Hh;
  if (i >= total) return;
  int h0 = (int)(i % Hh);
  long e = i / Hh;
  int s, d; gat_endpoints(ei, E, e, s, d);
  float v = gat_leaky(ss[(size_t)s * Hh + h0] + sd[(size_t)d * Hh + h0]);
  gat_atomic_max_f32(&emax[(size_t)d * Hh + h0], v);
}

__global__ void gat_edge_sum(const int* __restrict__ ei, int E, int N, int Hh,
                             const float* __restrict__ ss, const float* __restrict__ sd,
                             const float* __restrict__ emax, float* __restrict__ denom) {
  long i = (long)blockIdx.x * blockDim.x + threadIdx.x;
  long total = (long)(E + N) * Hh;
  if (i >= total) return;
  int h0 = (int)(i % Hh);
  long e = i / Hh;
  int s, d; gat_endpoints(ei, E, e, s, d);
  float v = gat_leaky(ss[(size_t)s * Hh + h0] + sd[(size_t)d * Hh + h0]);
  float ee = __expf(v - emax[(size_t)d * Hh + h0]);
  atomicAdd(&denom[(size_t)d * Hh + h0], ee);
}

// One wave32 per (edge, head); lanes cover F=64 features (2 per lane).
// Recomputes alpha (cheap, L2-resident scalars) to avoid a 13.6MB ee buffer.
__global__ void gat_edge_agg(const int* __restrict__ ei, int E, int N, int Hh, int F,
                             const float* __restrict__ ss, const float* __restrict__ sd,
                             const float* __restrict__ emax, const float* __restrict__ denom,
                             const float* __restrict__ feat, float* __restrict__ agg) {
  long gid  = (long)blockIdx.x * blockDim.x + threadIdx.x;
  long wave = gid >> 5;
  int  lane = (int)(gid & 31);
  long total = (long)(E + N) * Hh;
  if (wave >= total) return;
  int h0 = (int)(wave % Hh);
  long e = wave / Hh;
  int s, d; gat_endpoints(ei, E, e, s, d);
  float v = gat_leaky(ss[(size_t)s * Hh + h0] + sd[(size_t)d * Hh + h0]);
  float alpha = __expf(v - emax[(size_t)d * Hh + h0]) / denom[(size_t)d * Hh + h0];
  const float* hs = feat + ((size_t)s * Hh + h0) * F;
  float* od       = agg  + ((size_t)d * Hh + h0) * F;
  for (int f = lane; f < F; f += 32)
    atomicAdd(&od[f], hs[f] * alpha);
}

// ---------------------------------------------------------------------------
// Host-side launch
// ---------------------------------------------------------------------------
static inline unsigned cdivl(long a, long b) { return (unsigned)((a + b - 1) / b); }

extern "C" void kernel_launch(void* const* d_in, const int* in_sizes, int n_in,
                              void* d_out, int out_size, void* d_ws, size_t ws_size,
                              hipStream_t stream) {
  const float* x      = (const float*)d_in[0];
  const int*   ei     = (const int*)  d_in[1];
  const float* W1     = (const float*)d_in[2];
  const float* a_src1 = (const float*)d_in[3];
  const float* a_dst1 = (const float*)d_in[4];
  const float* b1     = (const float*)d_in[5];
  const float* W2     = (const float*)d_in[6];
  const float* a_src2 = (const float*)d_in[7];
  const float* a_dst2 = (const float*)d_in[8];
  const float* b2     = (const float*)d_in[9];
  float* out = (float*)d_out;

  const int IN_ = 256, H_ = 4, HID_ = 64, OUT_ = 64;
  const int C1  = H_ * HID_;                 // 256
  const int Nn  = in_sizes[0] / IN_;         // 50000
  const int Ee  = in_sizes[1] / 2;           // 800000

  // Workspace layout (floats): ~105.6 MB total
  float* ws    = (float*)d_ws;
  float* h     = ws;                                  // [N, 256] projected feats (reused L2: [N,64])
  float* agg1  = h    + (size_t)Nn * C1;              // [N, 256] z1 -> x1 in place
  float* ssrc  = agg1 + (size_t)Nn * C1;              // [N, H]
  float* sdst  = ssrc + (size_t)Nn * H_;              // [N, H]
  float* emax  = sdst + (size_t)Nn * H_;              // [N, H]
  float* denom = emax + (size_t)Nn * H_;              // [N, H]

  const int TB = 256;
  const long NH1  = (long)Nn * H_;
  const long NC1  = (long)Nn * C1;
  const long tot1 = (long)(Ee + Nn) * H_;   // (edge,head) items, layer 1
  const long tot2 = (long)(Ee + Nn);        // layer 2 (1 head)
  const long NO   = (long)Nn * OUT_;
  const unsigned mblk = cdivl(Nn / 16, 8);  // 8 wave-strips of 16 rows per block

  // ---------------- Layer 1: GAT(4 heads, concat) ----------------
  gat_gemm_wmma<<<dim3(C1 / 64, mblk), 256, 0, stream>>>(x, W1, h, Nn, IN_, C1);
  gat_logits<<<cdivl(NH1, TB), TB, 0, stream>>>(h, a_src1, a_dst1, ssrc, sdst, NH1, H_, HID_);
  gat_fill<<<cdivl(NH1, TB), TB, 0, stream>>>(emax, NH1, -INFINITY);
  gat_fill<<<cdivl(NH1, TB), TB, 0, stream>>>(denom, NH1, 0.0f);
  gat_bias_init<<<cdivl(NC1, TB), TB, 0, stream>>>(agg1, NC1, C1, b1);

  gat_edge_max<<<cdivl(tot1, TB), TB, 0, stream>>>(ei, Ee, Nn, H_, ssrc, sdst, emax);
  gat_edge_sum<<<cdivl(tot1, TB), TB, 0, stream>>>(ei, Ee, Nn, H_, ssrc, sdst, emax, denom);
  gat_edge_agg<<<cdivl(tot1 * 32, TB), TB, 0, stream>>>(ei, Ee, Nn, H_, HID_,
                                                        ssrc, sdst, emax, denom, h, agg1);
  // z1 -> x1 = BETA*z1 + (C-BETA)*swish(z1), in place
  gat_mix<<<cdivl(NC1, TB), TB, 0, stream>>>(agg1, NC1);

  // ---------------- Layer 2: GAT(1 head, mean) -------------------
  gat_gemm_wmma<<<dim3(OUT_ / 64, mblk), 256, 0, stream>>>(agg1, W2, h, Nn, C1, OUT_);
  gat_logits<<<cdivl((long)Nn, TB), TB, 0, stream>>>(h, a_src2, a_dst2, ssrc, sdst, Nn, 1, OUT_);
  gat_fill<<<cdivl((long)Nn, TB), TB, 0, stream>>>(emax, Nn, -INFINITY);
  gat_fill<<<cdivl((long)Nn, TB), TB, 0, stream>>>(denom, Nn, 0.0f);
  gat_bias_init<<<cdivl(NO, TB), TB, 0, stream>>>(out, NO, OUT_, b2);

  gat_edge_max<<<cdivl(tot2, TB), TB, 0, stream>>>(ei, Ee, Nn, 1, ssrc, sdst, emax);
  gat_edge_sum<<<cdivl(tot2, TB), TB, 0, stream>>>(ei, Ee, Nn, 1, ssrc, sdst, emax, denom);
  gat_edge_agg<<<cdivl(tot2 * 32, TB), TB, 0, stream>>>(ei, Ee, Nn, 1, OUT_,
                                                        ssrc, sdst, emax, denom, h, out);
  gat_scale<<<cdivl(NO, TB), TB, 0, stream>>>(out, NO, CCONST);
}